// MutualInformation_myfunc_39513699123813
// MI455X (gfx1250) — compile-verified
//
#include <hip/hip_runtime.h>
#include <math.h>

typedef __attribute__((ext_vector_type(2))) float v2f;
typedef __attribute__((ext_vector_type(8))) float v8f;

#define NROWS 8192
#define DF 64
#define DZ 32
#define KF 128              // 2*DF (concat [mean^2, -2*mean] x [iv, s*iv])
#define KZ 64               // 2*DZ
#define SPLIT 8             // column splits per 16-row strip
#define JCHUNK (NROWS / SPLIT)
#define WAVES_PER_BLOCK 4
#define NUM_TRAIN 60000
#define LOG2PI 1.8378770664093453f

// ---- workspace layout (float offsets) ----
#define OFF_AF 0
#define OFF_BF (OFF_AF + NROWS * KF)
#define OFF_AZ (OFF_BF + NROWS * KF)
#define OFF_BZ (OFF_AZ + NROWS * KZ)
#define OFF_CF (OFF_BZ + NROWS * KZ)
#define OFF_CZ (OFF_CF + NROWS)
#define OFF_PART (OFF_CZ + NROWS)                    // 3*NROWS*SPLIT*2
#define OFF_ROWLSE (OFF_PART + 3 * NROWS * SPLIT * 2) // 3*NROWS

// ---------------------------------------------------------------------------
// Kernel 1: build GEMM panels + per-column log-density constants.
//   Af[i,k] : k<DF -> mean^2 ; k>=DF -> -2*mean       (row-major, K contiguous)
//   Bf[j,k] : k<DF -> iv     ; k>=DF -> sample*iv
//   cf[j]   = -0.5*(DF*log2pi + sum(logvar) + sum(sample^2*iv))
// so  log_q[i,j] = cf[j] - 0.5 * dot_k(Af[i,:], Bf[j,:])
// ---------------------------------------------------------------------------
__global__ void mi_prep(const float* __restrict__ fm, const float* __restrict__ flv,
                        const float* __restrict__ fs, const float* __restrict__ zm,
                        const float* __restrict__ zlv, const float* __restrict__ zs,
                        float* __restrict__ Af, float* __restrict__ Bf,
                        float* __restrict__ Az, float* __restrict__ Bz,
                        float* __restrict__ cf, float* __restrict__ cz) {
  const int i = blockIdx.x * blockDim.x + threadIdx.x;
  if (i >= NROWS) return;

  float sumlv = 0.f, a = 0.f;
  for (int d = 0; d < DF; ++d) {
    const float lv = flv[i * DF + d];
    const float iv = __expf(-lv);
    const float s = fs[i * DF + d];
    const float m = fm[i * DF + d];
    Bf[i * KF + d] = iv;
    Bf[i * KF + DF + d] = s * iv;
    Af[i * KF + d] = m * m;
    Af[i * KF + DF + d] = -2.0f * m;
    sumlv += lv;
    a += s * s * iv;
  }
  cf[i] = -0.5f * ((float)DF * LOG2PI + sumlv + a);

  sumlv = 0.f; a = 0.f;
  for (int d = 0; d < DZ; ++d) {
    const float lv = zlv[i * DZ + d];
    const float iv = __expf(-lv);
    const float s = zs[i * DZ + d];
    const float m = zm[i * DZ + d];
    Bz[i * KZ + d] = iv;
    Bz[i * KZ + DZ + d] = s * iv;
    Az[i * KZ + d] = m * m;
    Az[i * KZ + DZ + d] = -2.0f * m;
    sumlv += lv;
    a += s * s * iv;
  }
  cz[i] = -0.5f * ((float)DZ * LOG2PI + sumlv + a);
}

// ---------------------------------------------------------------------------
// Kernel 2: fused dual-GEMM (fp32 WMMA 16x16x4) + online logsumexp.
// One wave = one 16-row strip x one 1/SPLIT column chunk. No N*N matrix is
// ever materialized; all three LSE statistics stream through registers.
// ---------------------------------------------------------------------------
__global__ __launch_bounds__(WAVES_PER_BLOCK * 32)
void mi_main(const float* __restrict__ Af, const float* __restrict__ Bf,
             const float* __restrict__ Az, const float* __restrict__ Bz,
             const float* __restrict__ cf, const float* __restrict__ cz,
             float* __restrict__ part) {
  const int lane = threadIdx.x & 31;
  const int wave = threadIdx.x >> 5;
  const int task = blockIdx.x * WAVES_PER_BLOCK + wave;
  const int strip = task / SPLIT;
  const int split = task % SPLIT;
  const int i0 = strip * 16;
  const int jbeg = split * JCHUNK;

  const int n = lane & 15;   // A: row index M ; B: col index N (ISA 7.12.2)
  const int h = lane >> 4;   // lane half selects K pair (K = 2h, 2h+1)

  // Register-resident A fragments for the whole column loop.
  v2f af[KF / 4];
  {
    const v2f* arow = (const v2f*)(Af + (size_t)(i0 + n) * KF);
#pragma unroll
    for (int kf = 0; kf < KF / 4; ++kf) af[kf] = arow[2 * kf + h];
  }
  v2f az[KZ / 4];
  {
    const v2f* arow = (const v2f*)(Az + (size_t)(i0 + n) * KZ);
#pragma unroll
    for (int kf = 0; kf < KZ / 4; ++kf) az[kf] = arow[2 * kf + h];
  }

  // Online LSE state: per lane, 8 rows (M = r + 8*h), 3 statistics.
  float mxf[8], smf[8], mxz[8], smz[8], mxfz[8], smfz[8];
#pragma unroll
  for (int r = 0; r < 8; ++r) {
    mxf[r] = -INFINITY; smf[r] = 0.f;
    mxz[r] = -INFINITY; smz[r] = 0.f;
    mxfz[r] = -INFINITY; smfz[r] = 0.f;
  }

  for (int j0 = jbeg; j0 < jbeg + JCHUNK; j0 += 16) {
    v8f accf = {0.f, 0.f, 0.f, 0.f, 0.f, 0.f, 0.f, 0.f};
    v8f accz = {0.f, 0.f, 0.f, 0.f, 0.f, 0.f, 0.f, 0.f};

    const v2f* bfrow = (const v2f*)(Bf + (size_t)(j0 + n) * KF);
#pragma unroll
    for (int kf = 0; kf < KF / 4; ++kf) {
      v2f b = bfrow[2 * kf + h];
      accf = __builtin_amdgcn_wmma_f32_16x16x4_f32(false, af[kf], false, b,
                                                   (short)0, accf, false, false);
    }
    const v2f* bzrow = (const v2f*)(Bz + (size_t)(j0 + n) * KZ);
#pragma unroll
    for (int kf = 0; kf < KZ / 4; ++kf) {
      v2f b = bzrow[2 * kf + h];
      accz = __builtin_amdgcn_wmma_f32_16x16x4_f32(false, az[kf], false, b,
                                                   (short)0, accz, false, false);
    }

    const float cfj = cf[j0 + n];
    const float czj = cz[j0 + n];
#pragma unroll
    for (int r = 0; r < 8; ++r) {
      const float lf = __builtin_fmaf(-0.5f, accf[r], cfj);
      const float lz = __builtin_fmaf(-0.5f, accz[r], czj);
      const float lfz = lf + lz;
      float nm;
      nm = fmaxf(mxf[r], lf);
      smf[r] = smf[r] * __expf(mxf[r] - nm) + __expf(lf - nm);
      mxf[r] = nm;
      nm = fmaxf(mxz[r], lz);
      smz[r] = smz[r] * __expf(mxz[r] - nm) + __expf(lz - nm);
      mxz[r] = nm;
      nm = fmaxf(mxfz[r], lfz);
      smfz[r] = smfz[r] * __expf(mxfz[r] - nm) + __expf(lfz - nm);
      mxfz[r] = nm;
    }
  }

  // Combine the 16 columns held across each 16-lane half (xor butterflies
  // with strides 1..8 stay inside a half in wave32).
#pragma unroll
  for (int r = 0; r < 8; ++r) {
#pragma unroll
    for (int d = 1; d < 16; d <<= 1) {
      float om, os, nm;
      om = __shfl_xor(mxf[r], d, 32); os = __shfl_xor(smf[r], d, 32);
      nm = fmaxf(mxf[r], om);
      smf[r] = smf[r] * __expf(mxf[r] - nm) + os * __expf(om - nm);
      mxf[r] = nm;
      om = __shfl_xor(mxz[r], d, 32); os = __shfl_xor(smz[r], d, 32);
      nm = fmaxf(mxz[r], om);
      smz[r] = smz[r] * __expf(mxz[r] - nm) + os * __expf(om - nm);
      mxz[r] = nm;
      om = __shfl_xor(mxfz[r], d, 32); os = __shfl_xor(smfz[r], d, 32);
      nm = fmaxf(mxfz[r], om);
      smfz[r] = smfz[r] * __expf(mxfz[r] - nm) + os * __expf(om - nm);
      mxfz[r] = nm;
    }
  }

  // Lane 0 of each half writes its 8 rows' (max, sum) partials.
  if (n == 0) {
#pragma unroll
    for (int r = 0; r < 8; ++r) {
      const int row = i0 + r + 8 * h;
      const size_t b0 = ((size_t)(0 * NROWS + row) * SPLIT + split) * 2;
      const size_t b1 = ((size_t)(1 * NROWS + row) * SPLIT + split) * 2;
      const size_t b2 = ((size_t)(2 * NROWS + row) * SPLIT + split) * 2;
      part[b0] = mxf[r];  part[b0 + 1] = smf[r];
      part[b1] = mxz[r];  part[b1 + 1] = smz[r];
      part[b2] = mxfz[r]; part[b2 + 1] = smfz[r];
    }
  }
}

// ---------------------------------------------------------------------------
// Kernel 3: merge SPLIT partials per row -> per-row logsumexp.
// ---------------------------------------------------------------------------
__global__ void mi_reduce_rows(const float* __restrict__ part,
                               float* __restrict__ rowlse) {
  const int row = blockIdx.x * blockDim.x + threadIdx.x;
  if (row >= NROWS) return;
#pragma unroll
  for (int stat = 0; stat < 3; ++stat) {
    float m = -INFINITY, s = 0.f;
    const float* p = part + (size_t)(stat * NROWS + row) * SPLIT * 2;
    for (int k = 0; k < SPLIT; ++k) {
      const float pm = p[2 * k], ps = p[2 * k + 1];
      const float nm = fmaxf(m, pm);
      s = s * __expf(m - nm) + ps * __expf(pm - nm);
      m = nm;
    }
    rowlse[stat * NROWS + row] = m + __logf(s);
  }
}

// ---------------------------------------------------------------------------
// Kernel 4: deterministic mean over rows + final entropies / MI.
// ---------------------------------------------------------------------------
__global__ void mi_final(const float* __restrict__ rowlse, float* __restrict__ out) {
  __shared__ float red0[256], red1[256], red2[256];
  const int tid = threadIdx.x;
  float s0 = 0.f, s1 = 0.f, s2 = 0.f;
  for (int row = tid; row < NROWS; row += 256) {
    s0 += rowlse[row];
    s1 += rowlse[NROWS + row];
    s2 += rowlse[2 * NROWS + row];
  }
  red0[tid] = s0; red1[tid] = s1; red2[tid] = s2;
  __syncthreads();
  for (int off = 128; off > 0; off >>= 1) {
    if (tid < off) {
      red0[tid] += red0[tid + off];
      red1[tid] += red1[tid + off];
      red2[tid] += red2[tid + off];
    }
    __syncthreads();
  }
  if (tid == 0) {
    const float logNM = logf((float)NUM_TRAIN * (float)NROWS);
    const float inv = 1.0f / (float)NROWS;
    const float Hf = logNM - red0[0] * inv;
    const float Hz = logNM - red1[0] * inv;
    const float Hfz = logNM - red2[0] * inv;
    float mi = Hf + Hz - Hfz;
    if (mi < 0.f) mi = 0.f;
    out[0] = Hf; out[1] = Hz; out[2] = Hfz; out[3] = mi;
  }
}

extern "C" void kernel_launch(void* const* d_in, const int* in_sizes, int n_in,
                              void* d_out, int out_size, void* d_ws, size_t ws_size,
                              hipStream_t stream) {
  const float* fm = (const float*)d_in[0];
  const float* flv = (const float*)d_in[1];
  const float* fs = (const float*)d_in[2];
  const float* zm = (const float*)d_in[3];
  const float* zlv = (const float*)d_in[4];
  const float* zs = (const float*)d_in[5];

  float* ws = (float*)d_ws;
  float* Af = ws + OFF_AF;
  float* Bf = ws + OFF_BF;
  float* Az = ws + OFF_AZ;
  float* Bz = ws + OFF_BZ;
  float* cf = ws + OFF_CF;
  float* cz = ws + OFF_CZ;
  float* part = ws + OFF_PART;
  float* rowlse = ws + OFF_ROWLSE;

  mi_prep<<<NROWS / 256, 256, 0, stream>>>(fm, flv, fs, zm, zlv, zs,
                                           Af, Bf, Az, Bz, cf, cz);

  const int tasks = (NROWS / 16) * SPLIT;                 // 4096 waves
  mi_main<<<tasks / WAVES_PER_BLOCK, WAVES_PER_BLOCK * 32, 0, stream>>>(
      Af, Bf, Az, Bz, cf, cz, part);

  mi_reduce_rows<<<NROWS / 256, 256, 0, stream>>>(part, rowlse);
  mi_final<<<1, 256, 0, stream>>>(rowlse, (float*)d_out);
}